// SoS_Loss_43190191128804
// MI455X (gfx1250) — compile-verified
//
#include <hip/hip_runtime.h>

#define B_DIM 128
#define S_DIM 32768
#define P_DIM 65536
#define POS_WEIGHT 5.0f

typedef __attribute__((ext_vector_type(2))) float v2f;
typedef __attribute__((ext_vector_type(8))) float v8f;

// ---------------------------------------------------------------------------
// Kernel 0: zero the gt byte-map (B*S bytes) with 128-bit stores.
// ---------------------------------------------------------------------------
__global__ void zero_gt_kernel(uint4* __restrict__ p, int n4) {
    int i = blockIdx.x * blockDim.x + threadIdx.x;
    if (i < n4) p[i] = make_uint4(0u, 0u, 0u, 0u);
}

// ---------------------------------------------------------------------------
// Kernel 1: scatter gt.
//   pj is sorted per batch -> is_first = (k==0) || (pj[k] != pj[k-1]).
//   Fetch (pi,pj) as one 128-bit load; get pj[k-1] from the previous lane via
//   wave32 shuffle (P_DIM % 32 == 0, so k==0 only ever occurs on lane 0).
// ---------------------------------------------------------------------------
__global__ void scatter_gt_kernel(const long long* __restrict__ paths,
                                  const int* __restrict__ targets,
                                  unsigned char* __restrict__ gt) {
    const long long g = (long long)blockIdx.x * blockDim.x + threadIdx.x; // 0..B*P-1
    const int k = (int)(g & (long long)(P_DIM - 1));
    const int b = (int)(g >> 16);

    const longlong2* pv = (const longlong2*)paths;
    longlong2 pp = pv[g];               // global_load_b128, fully coalesced
    const int pi = (int)pp.x;
    const int pj = (int)pp.y;

    const int lane = threadIdx.x & 31;
    int prev = __shfl_up(pj, 1, 32);    // pj of step k-1 for lanes 1..31
    if (lane == 0) {
        prev = (k > 0) ? (int)paths[2 * g - 1] : -1;
    }
    const bool is_first = (k == 0) || (pj != prev);

    if (is_first && targets[(size_t)b * S_DIM + pj] == 1) {
        gt[(size_t)b * S_DIM + pi] = (unsigned char)1; // races write identical 1
    }
}

// ---------------------------------------------------------------------------
// Kernel 2: per-element weighted-BCE loss, accumulated with
// V_WMMA_F32_16X16X4_F32 (A = ones(16x4)) -> D[m,n] = sum_k B[k,n] + C[m,n].
// Each wave consumes 64 contiguous elements per iteration (2 per lane).
// The chunk counter goes through readfirstlane so the loop is provably
// wave-uniform: scalar s_cmp/s_cbranch, EXEC stays all-ones across every WMMA
// (ISA requirement for V_WMMA_*).
// ---------------------------------------------------------------------------
__device__ __forceinline__ float loss_elem(float p, float g) {
    // log_sigmoid(x) = min(x,0) - log1p(exp(-|x|))
    const float t = log1pf(__expf(-fabsf(p)));
    const float ls_p  = fminf(p, 0.0f)  - t;   // log_sigmoid(p)
    const float ls_np = fminf(-p, 0.0f) - t;   // log_sigmoid(-p)
    return -(POS_WEIGHT * g * ls_p + (1.0f - g) * ls_np);
}

__global__ void loss_wmma_kernel(const float* __restrict__ preds,
                                 const unsigned char* __restrict__ gt,
                                 float* __restrict__ partials) {
    const int lane = threadIdx.x & 31;
    // force scalar (SGPR) wave id -> uniform loop, EXEC untouched
    const int wid =
        __builtin_amdgcn_readfirstlane((blockIdx.x * blockDim.x + threadIdx.x) >> 5);
    const int nwaves  = (gridDim.x * blockDim.x) >> 5;     // already uniform
    const int nchunks = (B_DIM * S_DIM) / 64;              // 65536, divides exactly

    v8f acc  = {0.f, 0.f, 0.f, 0.f, 0.f, 0.f, 0.f, 0.f};
    v2f ones = {1.0f, 1.0f};

    for (int c = wid; c < nchunks; c += nwaves) {
        const int base = c * 64 + lane * 2;
        v2f p = *(const v2f*)(preds + base);                       // 8B coalesced
        const unsigned int gpair = *(const unsigned short*)(gt + base); // 2B coalesced
        const float g0 = (float)(gpair & 0xFFu);
        const float g1 = (float)(gpair >> 8);
        v2f vals;
        vals.x = loss_elem(p.x, g0);
        vals.y = loss_elem(p.y, g1);
        // D = ones(16x4) x B(4x16) + C : exact f32 accumulation of 64 values
        acc = __builtin_amdgcn_wmma_f32_16x16x4_f32(
            false, ones, false, vals, (short)0, acc, false, false);
    }

    // acc[0]: lanes 0-15 = colsum[0..15] (M=0), lanes 16-31 = colsum[0..15] (M=8)
    float s = acc[0];
#pragma unroll
    for (int m = 16; m >= 1; m >>= 1) s += __shfl_xor(s, m, 32);
    if (lane == 0) partials[wid] = s * 0.5f;   // both halves are duplicates
}

// ---------------------------------------------------------------------------
// Kernel 3: deterministic fixed-order reduction of the per-wave partials.
// ---------------------------------------------------------------------------
__global__ void final_reduce_kernel(const float* __restrict__ partials, int n,
                                    float* __restrict__ out) {
    __shared__ float sm[256];
    float s = 0.0f;
    for (int i = threadIdx.x; i < n; i += 256) s += partials[i];
    sm[threadIdx.x] = s;
    __syncthreads();
    for (int off = 128; off >= 1; off >>= 1) {
        if ((int)threadIdx.x < off) sm[threadIdx.x] += sm[threadIdx.x + off];
        __syncthreads();
    }
    if (threadIdx.x == 0)
        out[0] = sm[0] * (1.0f / ((float)B_DIM * (float)S_DIM));
}

// ---------------------------------------------------------------------------
extern "C" void kernel_launch(void* const* d_in, const int* in_sizes, int n_in,
                              void* d_out, int out_size, void* d_ws, size_t ws_size,
                              hipStream_t stream) {
    const float*     preds   = (const float*)d_in[0];
    const int*       targets = (const int*)d_in[1];
    const long long* paths   = (const long long*)d_in[2];

    unsigned char* gt = (unsigned char*)d_ws;                          // B*S bytes = 4 MiB
    float* partials   = (float*)((char*)d_ws + (size_t)B_DIM * S_DIM); // after gt

    // 0) zero gt
    const int n4 = (B_DIM * S_DIM) / 16;       // 262144 uint4
    zero_gt_kernel<<<n4 / 256, 256, 0, stream>>>((uint4*)gt, n4);

    // 1) scatter hits into gt  (B*P threads, exact)
    const long long nsteps = (long long)B_DIM * P_DIM;  // 8,388,608
    scatter_gt_kernel<<<(unsigned)(nsteps / 256), 256, 0, stream>>>(paths, targets, gt);

    // 2) loss + WMMA accumulation
    const int LBLOCKS = 512, LTHREADS = 256;
    const int nwaves = LBLOCKS * LTHREADS / 32;  // 4096 partials (16 KiB)
    loss_wmma_kernel<<<LBLOCKS, LTHREADS, 0, stream>>>(preds, gt, partials);

    // 3) deterministic final reduce
    final_reduce_kernel<<<1, 256, 0, stream>>>(partials, nwaves, (float*)d_out);
}